// ListwiseRegressionLoss_57767310131696
// MI455X (gfx1250) — compile-verified
//
#include <hip/hip_runtime.h>
#include <hip/hip_bf16.h>

// ListwiseRegressionLoss on MI455X (gfx1250).
//
// Math: lse - s = log(rank+1) exactly in fp32 (exp(-1e9) underflows to 0),
// so result = sum_g lgamma(n_g + 1) / #nonempty_groups, where n_g are the
// group sizes. y_pred / y_true cancel. group_ids is sorted, so group sizes
// come from segment boundaries: no sort, no atomics, 16 MB of traffic total
// (~0.7 us at 23.3 TB/s HBM; fits in the 192 MB L2 anyway).
//
// CDNA5-specific paths used:
//  * global_load_async_to_lds_b128 + s_wait_asynccnt (async data mover path)
//  * v_wmma_f32_16x16x4_f32 for the final dual cross-lane reduction
//    (loss total in columns 0-7, unique-group count in columns 8-15).

#define NUM_GROUPS 2048
#define TILE_INTS  1024   // 256 threads x 4 ints, one b128 async per thread

typedef __attribute__((ext_vector_type(2))) float v2f;
typedef __attribute__((ext_vector_type(8))) float v8f;

__global__ void lw_init_kernel(int* __restrict__ p, int n) {
    int i = blockIdx.x * blockDim.x + threadIdx.x;
    if (i < n) p[i] = 0;
}

// Detect segment boundaries in the sorted group_ids array.
// starts[g] = first index of group g, ends[g] = one past last index.
// Empty groups keep starts==ends==0 from the init kernel.
__global__ __launch_bounds__(256)
void lw_boundary_kernel(const int* __restrict__ gids,
                        int* __restrict__ starts,
                        int* __restrict__ ends,
                        int N) {
    __shared__ int tile[TILE_INTS];
    const int t  = threadIdx.x;
    const int p0 = blockIdx.x * TILE_INTS + t * 4;

    // Stage this block's 4 KB tile into LDS via the CDNA5 async path.
    // Each lane copies its own 16 bytes: LDS address taken from vdst; per ISA
    // flat/LDS addressing the low 32 bits of a generic LDS pointer are the
    // LDS byte offset.
    if (p0 + 4 <= N) {
        unsigned ldsOff = (unsigned)(size_t)(&tile[t * 4]);
        unsigned long long gaddr = (unsigned long long)(size_t)(gids + p0);
        asm volatile("global_load_async_to_lds_b128 %0, %1, off"
                     :
                     : "v"(ldsOff), "v"(gaddr)
                     : "memory");
    } else {
        #pragma unroll
        for (int j = 0; j < 4; ++j)
            if (p0 + j < N) tile[t * 4 + j] = gids[p0 + j];
    }
    asm volatile("s_wait_asynccnt 0" ::: "memory");
    __syncthreads();

    if (p0 >= N) return;

    int vals[5];
    #pragma unroll
    for (int j = 0; j < 4; ++j) vals[j] = tile[t * 4 + j];
    // One element past this thread's quad (for the j==3 transition check).
    if (t < 255)              vals[4] = tile[t * 4 + 4];
    else if (p0 + 4 < N)      vals[4] = gids[p0 + 4];
    else                      vals[4] = vals[3];

    if (p0 == 0) starts[vals[0]] = 0;   // first segment start

    #pragma unroll
    for (int j = 0; j < 4; ++j) {
        int q = p0 + j + 1;             // boundary between q-1 and q
        if (q > N) break;
        int a = vals[j];
        if (q == N) {
            ends[a] = N;                // final segment end
        } else {
            int b = vals[j + 1];
            if (a != b) {               // ~2048 transitions total: cheap stores
                ends[a]   = q;
                starts[b] = q;
            }
        }
    }
}

// Single-wave finalize: per-lane partial (loss, count) over 64 groups each,
// then one v_wmma_f32_16x16x4_f32 reduces both across the wave.
// A (16x4): K0/K1 slots = per-lane loss partials, K2/K3 = count partials.
// B (4x16): 1 where (K<2 && N<8) || (K>=2 && N>=8), so columns 0-7 of D
// accumulate loss and columns 8-15 accumulate count; one shfl_xor(16) folds
// the two row halves of D.
__global__ __launch_bounds__(32)
void lw_finalize_kernel(const int* __restrict__ starts,
                        const int* __restrict__ ends,
                        float* __restrict__ out) {
    const int lane = threadIdx.x;
    float lsum = 0.0f, lcnt = 0.0f;
    for (int g = lane; g < NUM_GROUPS; g += 32) {
        int n = ends[g] - starts[g];
        if (n > 0) {
            lsum += lgammaf((float)n + 1.0f);   // log(n!)
            lcnt += 1.0f;
        }
    }

    float lsum_o = __shfl_xor(lsum, 16, 32);
    float lcnt_o = __shfl_xor(lcnt, 16, 32);

    v2f a;
    a.x = (lane < 16) ? lsum   : lcnt_o;   // A[row][0] / A[row][2]
    a.y = (lane < 16) ? lsum_o : lcnt;     // A[row][1] / A[row][3]

    float bv = ((lane < 8) || (lane >= 24)) ? 1.0f : 0.0f;
    v2f b;  b.x = bv;  b.y = bv;

    v8f c = {};
    v8f d = __builtin_amdgcn_wmma_f32_16x16x4_f32(
        /*neg_a=*/false, a, /*neg_b=*/false, b,
        /*c_mod=*/(short)0, c, /*reuse_a=*/false, /*reuse_b=*/false);

    float e = d[0] + d[1] + d[2] + d[3] + d[4] + d[5] + d[6] + d[7];
    float f = e + __shfl_xor(e, 16, 32);   // cols<8: loss total; cols>=8: count
    float cnt = __shfl(f, 8, 32);
    if (lane == 0) out[0] = f / cnt;
}

extern "C" void kernel_launch(void* const* d_in, const int* in_sizes, int n_in,
                              void* d_out, int out_size, void* d_ws, size_t ws_size,
                              hipStream_t stream) {
    // Inputs: y_pred (f32), y_true (f32), group_ids (i32).
    // y_pred / y_true cancel analytically (see header comment).
    (void)d_in[0]; (void)d_in[1]; (void)n_in; (void)out_size; (void)ws_size;
    const int* gids = (const int*)d_in[2];
    const int  N    = in_sizes[2];

    int* starts = (int*)d_ws;              // NUM_GROUPS ints
    int* ends   = starts + NUM_GROUPS;     // NUM_GROUPS ints

    lw_init_kernel<<<(2 * NUM_GROUPS + 255) / 256, 256, 0, stream>>>(starts, 2 * NUM_GROUPS);

    int nblocks = (N + TILE_INTS - 1) / TILE_INTS;   // 4096 blocks for N = 4M
    lw_boundary_kernel<<<nblocks, 256, 0, stream>>>(gids, starts, ends, N);

    lw_finalize_kernel<<<1, 32, 0, stream>>>(starts, ends, (float*)d_out);
}